// GatedGraphNeuralNetwork_85856396247056
// MI455X (gfx1250) — compile-verified
//
#include <hip/hip_runtime.h>
#include <cstddef>

#define H 128
#define A_DIM 16
#define NUM_ET 4
#define STEPS 4
#define N_NODES 100000
#define E_PER_TYPE 250000

typedef float v2f __attribute__((ext_vector_type(2)));
typedef float v8f __attribute__((ext_vector_type(8)));

// D = A(16x4) * B(4x16) + C(16x16), full f32 precision on the matrix pipe.
__device__ __forceinline__ v8f wmma4(v2f a, v2f b, v8f c) {
    return __builtin_amdgcn_wmma_f32_16x16x4_f32(false, a, false, b, (short)0, c, false, false);
}

__device__ __forceinline__ float fsigmoid(float x) {
    return 1.0f / (1.0f + __expf(-x));
}

// ---------------------------------------------------------------------------
// Pack a (K x N) B-matrix into WMMA B-operand order.
// packed float2[(kk*(N/16)+nt)*32 + lane] = { B[k0][n], B[k0+1][n] }
// with n = nt*16 + (lane&15), k0 = kk*4 + 2*(lane>>4).
// transposed==1: src is (N x K) row-major, B[k][n] = src[n*K + k].
// ---------------------------------------------------------------------------
__global__ void pack_b_kernel(const float* __restrict__ src, float* __restrict__ dst,
                              int K, int N, int transposed) {
    int total = (K / 4) * (N / 16) * 32;
    int stride = gridDim.x * blockDim.x;
    for (int p = blockIdx.x * blockDim.x + threadIdx.x; p < total; p += stride) {
        int lane = p & 31;
        int tmp = p >> 5;
        int ntiles = N >> 4;
        int nt = tmp % ntiles;
        int kk = tmp / ntiles;
        int nn = lane & 15;
        int koff = (lane >> 4) * 2;
        int k0 = kk * 4 + koff;
        int n = nt * 16 + nn;
        float b0, b1;
        if (transposed) {
            b0 = src[(size_t)n * K + k0];
            b1 = src[(size_t)n * K + k0 + 1];
        } else {
            b0 = src[(size_t)k0 * N + n];
            b1 = src[(size_t)(k0 + 1) * N + n];
        }
        dst[2 * p + 0] = b0;
        dst[2 * p + 1] = b1;
    }
}

__global__ void zero_kernel(float4* __restrict__ p, int n4) {
    int stride = gridDim.x * blockDim.x;
    for (int i = blockIdx.x * blockDim.x + threadIdx.x; i < n4; i += stride)
        p[i] = make_float4(0.f, 0.f, 0.f, 0.f);
}

// ---------------------------------------------------------------------------
// h0 = concat(x, ann) @ W_hidden + b_hidden.  One wave per 16-node tile.
// K = 144 (36 k-steps), N = 128 (8 n-tiles).  Runs once; global B path is fine.
// ---------------------------------------------------------------------------
__global__ void init_h_kernel(const float* __restrict__ x, const float* __restrict__ ann,
                              const float* __restrict__ pW, const float* __restrict__ b_hidden,
                              float* __restrict__ hout, int nTiles) {
    int wave = (blockIdx.x * blockDim.x + threadIdx.x) >> 5;
    int lane = threadIdx.x & 31;
    if (wave >= nTiles) return;                 // wave-uniform: EXEC stays all-ones
    int tile = wave;
    int row = tile * 16 + (lane & 15);
    int koff = (lane >> 4) * 2;
    const float2* pW2 = (const float2*)pW;

    v8f acc[8] = {};
    for (int kk = 0; kk < 36; ++kk) {
        int k0 = kk * 4 + koff;
        v2f a;
        if (k0 < H) {
            const float2 ax = *(const float2*)(x + (size_t)row * H + k0);
            a.x = ax.x; a.y = ax.y;
        } else {
            const float2 aa = *(const float2*)(ann + (size_t)row * A_DIM + (k0 - H));
            a.x = aa.x; a.y = aa.y;
        }
#pragma unroll
        for (int nt = 0; nt < 8; ++nt) {
            float2 bp = pW2[(kk * 8 + nt) * 32 + lane];
            v2f b; b.x = bp.x; b.y = bp.y;
            acc[nt] = wmma4(a, b, acc[nt]);
        }
    }
    int mbase = tile * 16 + 8 * (lane >> 4);
    int nn = lane & 15;
#pragma unroll
    for (int nt = 0; nt < 8; ++nt) {
        int n = nt * 16 + nn;
        float bias = b_hidden[n];
#pragma unroll
        for (int r = 0; r < 8; ++r)
            hout[(size_t)(mbase + r) * H + n] = acc[nt][r] + bias;
    }
}

// ---------------------------------------------------------------------------
// Messages + scatter.  Block = 25 waves (800 thr), one wave per 16-edge group.
// blockIdx.y = edge type.  Packed W_msg[t] (64 KB) staged in LDS once per
// block; B operands served by ds_load_b64 (conflict-free: 8B lane stride).
// Grid is exact: 625 blocks * 25 waves = 15625 groups = E/16.
// ---------------------------------------------------------------------------
__global__ void msg_kernel(const float* __restrict__ h, const int* __restrict__ edges,
                           const float* __restrict__ pWmsg, const float* __restrict__ b_msg,
                           float* __restrict__ incoming) {
    __shared__ float2 ldsW[H * H / 2];          // 64 KB
    int t = blockIdx.y;
    int tid = threadIdx.x;

    const float2* srcW = (const float2*)(pWmsg + (size_t)t * H * H);
    for (int i = tid; i < H * H / 2; i += blockDim.x)
        ldsW[i] = srcW[i];
    __syncthreads();

    int wave = blockIdx.x * (blockDim.x >> 5) + (tid >> 5);
    int lane = tid & 31;
    int group = wave;                            // 0 .. 15624, always valid
    const int* etab = edges + (size_t)t * E_PER_TYPE * 2;

    int srcNode = etab[(size_t)(group * 16 + (lane & 15)) * 2 + 0];
    int khalf = lane >> 4;
    const float2* arow = (const float2*)(h + (size_t)srcNode * H);

    v8f acc[8] = {};
    for (int kk = 0; kk < 32; ++kk) {
        float2 ap = arow[kk * 2 + khalf];
        v2f a; a.x = ap.x; a.y = ap.y;
#pragma unroll
        for (int nt = 0; nt < 8; ++nt) {
            float2 bp = ldsW[(kk * 8 + nt) * 32 + lane];
            v2f b; b.x = bp.x; b.y = bp.y;
            acc[nt] = wmma4(a, b, acc[nt]);
        }
    }

    // scatter: value(vgpr r, lane) = C[m = r + 8*(lane>>4)][n = nt*16 + (lane&15)]
    int mhalf = 8 * (lane >> 4);
    int nn = lane & 15;
    int tg[8];
#pragma unroll
    for (int r = 0; r < 8; ++r)
        tg[r] = etab[(size_t)(group * 16 + mhalf + r) * 2 + 1];

#pragma unroll
    for (int nt = 0; nt < 8; ++nt) {
        int n = nt * 16 + nn;
        float bias = b_msg[t * H + n];
#pragma unroll
        for (int r = 0; r < 8; ++r) {
            unsafeAtomicAdd(&incoming[(size_t)tg[r] * H + n], acc[nt][r] + bias);
        }
    }
}

// ---------------------------------------------------------------------------
// GRU cell, fused, in-place on h.  Block = 10 waves (320 thr), one wave per
// 16-node tile.  Grid exact: 625 blocks * 10 = 6250 tiles = N/16.
// Per (gate, sub): stage the 32 KB chunk of each packed W in LDS (64 KB
// total), barrier, then 32 k-steps x 4 n-tiles x 2 matrices of WMMA.
// All reads of old h complete before any write (one wave owns its rows).
// ---------------------------------------------------------------------------
__global__ void gru_kernel(float* __restrict__ h, const float* __restrict__ incoming,
                           const float* __restrict__ pWih, const float* __restrict__ pWhh,
                           const float* __restrict__ b_ih, const float* __restrict__ b_hh) {
    __shared__ float2 ldsI[4096];               // 32 KB: chunk of packed W_ih
    __shared__ float2 ldsH[4096];               // 32 KB: chunk of packed W_hh
    int tid = threadIdx.x;
    int wave = blockIdx.x * (blockDim.x >> 5) + (tid >> 5);
    int lane = tid & 31;
    int tile = wave;                             // 0 .. 6249, always valid
    int row = tile * 16 + (lane & 15);
    int khalf = lane >> 4;
    int nn = lane & 15;
    int mbase = tile * 16 + 8 * (lane >> 4);

    const float2* aI = (const float2*)(incoming + (size_t)row * H);
    const float2* aH = (const float2*)(h + (size_t)row * H);
    const float2* BI = (const float2*)pWih;
    const float2* BH = (const float2*)pWhh;

    v8f rg[8];   // gate r; reused to hold n-gate values in chunk 2
    v8f zg[8];   // gate z

    for (int gate = 0; gate < 3; ++gate) {
        for (int sub = 0; sub < 2; ++sub) {
            int ntBase = gate * 8 + sub * 4;    // n-tile index in [0,24)

            __syncthreads();                    // previous chunk fully consumed
            for (int i = tid; i < 4096; i += blockDim.x) {
                int l2 = i & 31;
                int j2 = (i >> 5) & 3;
                int k2 = i >> 7;                // 0..31
                ldsI[i] = BI[(k2 * 24 + ntBase + j2) * 32 + l2];
                ldsH[i] = BH[(k2 * 24 + ntBase + j2) * 32 + l2];
            }
            __syncthreads();

            v8f accI[4] = {};
            v8f accH[4] = {};
            for (int kk = 0; kk < 32; ++kk) {
                float2 fI = aI[kk * 2 + khalf];
                float2 fH = aH[kk * 2 + khalf];
                v2f ai; ai.x = fI.x; ai.y = fI.y;
                v2f ah; ah.x = fH.x; ah.y = fH.y;
#pragma unroll
                for (int j = 0; j < 4; ++j) {
                    float2 bi = ldsI[(kk * 4 + j) * 32 + lane];
                    float2 bh = ldsH[(kk * 4 + j) * 32 + lane];
                    v2f vb1; vb1.x = bi.x; vb1.y = bi.y;
                    v2f vb2; vb2.x = bh.x; vb2.y = bh.y;
                    accI[j] = wmma4(ai, vb1, accI[j]);
                    accH[j] = wmma4(ah, vb2, accH[j]);
                }
            }
#pragma unroll
            for (int j = 0; j < 4; ++j) {
                int n = (ntBase + j) * 16 + nn;    // col in [0,384)
                float bi = b_ih[n];
                float bh = b_hh[n];
                int jj = sub * 4 + j;              // 0..7 within gate chunk
                if (gate == 0) {
#pragma unroll
                    for (int r = 0; r < 8; ++r)
                        rg[jj][r] = fsigmoid(accI[j][r] + bi + accH[j][r] + bh);
                } else if (gate == 1) {
#pragma unroll
                    for (int r = 0; r < 8; ++r)
                        zg[jj][r] = fsigmoid(accI[j][r] + bi + accH[j][r] + bh);
                } else {
#pragma unroll
                    for (int r = 0; r < 8; ++r)
                        rg[jj][r] = tanhf(accI[j][r] + bi + rg[jj][r] * (accH[j][r] + bh));
                }
            }
        }
    }

    // final blend AFTER all reads of old h: h = (1-z)*n + z*h
#pragma unroll
    for (int jj = 0; jj < 8; ++jj) {
        int ncol = jj * 16 + nn;
#pragma unroll
        for (int r = 0; r < 8; ++r) {
            size_t idx = (size_t)(mbase + r) * H + ncol;
            float hv = h[idx];
            h[idx] = (1.f - zg[jj][r]) * rg[jj][r] + zg[jj][r] * hv;
        }
    }
}

// ---------------------------------------------------------------------------
extern "C" void kernel_launch(void* const* d_in, const int* in_sizes, int n_in,
                              void* d_out, int out_size, void* d_ws, size_t ws_size,
                              hipStream_t stream) {
    const float* x        = (const float*)d_in[0];
    const float* ann      = (const float*)d_in[1];
    const int*   edges    = (const int*)d_in[2];
    const float* W_hidden = (const float*)d_in[3];
    const float* b_hidden = (const float*)d_in[4];
    const float* W_msg    = (const float*)d_in[5];
    const float* b_msg    = (const float*)d_in[6];
    const float* W_ih     = (const float*)d_in[7];
    const float* W_hh     = (const float*)d_in[8];
    const float* b_ih     = (const float*)d_in[9];
    const float* b_hh     = (const float*)d_in[10];

    float* h = (float*)d_out;                    // h lives in d_out throughout

    float* ws       = (float*)d_ws;
    float* incoming = ws;                                         // N*H floats
    float* pWhid    = incoming + (size_t)N_NODES * H;             // 144*128
    float* pWmsg    = pWhid + (size_t)(H + A_DIM) * H;            // 4*128*128
    float* pWih     = pWmsg + (size_t)NUM_ET * H * H;             // 128*384
    float* pWhh     = pWih + (size_t)H * 3 * H;                   // 128*384

    const int nNodeTiles = N_NODES / 16;        // 6250

    // Pack weight matrices into WMMA B-operand layout (tiny, once per launch).
    pack_b_kernel<<<36, 256, 0, stream>>>(W_hidden, pWhid, H + A_DIM, H, 0);
    for (int t = 0; t < NUM_ET; ++t)
        pack_b_kernel<<<32, 256, 0, stream>>>(W_msg + (size_t)t * H * H,
                                              pWmsg + (size_t)t * H * H, H, H, 0);
    pack_b_kernel<<<96, 256, 0, stream>>>(W_ih, pWih, H, 3 * H, 1);
    pack_b_kernel<<<96, 256, 0, stream>>>(W_hh, pWhh, H, 3 * H, 1);

    // Initial projection (8 waves x 782 blocks, wave-uniform guard).
    init_h_kernel<<<782, 256, 0, stream>>>(x, ann, pWhid, b_hidden, h, nNodeTiles);

    for (int step = 0; step < STEPS; ++step) {
        zero_kernel<<<2048, 256, 0, stream>>>((float4*)incoming, (N_NODES * H) / 4);
        // 625 * 25 waves = 15625 edge groups exactly; y = edge type
        msg_kernel<<<dim3(625, NUM_ET), 800, 0, stream>>>(h, edges, pWmsg, b_msg, incoming);
        // 625 * 10 waves = 6250 node tiles exactly
        gru_kernel<<<625, 320, 0, stream>>>(h, incoming, pWih, pWhh, b_ih, b_hh);
    }
}